// FFTConv_27204322853342
// MI455X (gfx1250) — compile-verified
//
#include <hip/hip_runtime.h>
#include <hip/hip_bf16.h>

// CDNA5 (gfx1250) wave32 WMMA types
typedef __attribute__((ext_vector_type(2))) float v2f;
typedef __attribute__((ext_vector_type(8))) float v8f;

__device__ __forceinline__ v8f wmma4(v2f a, v2f b, v8f c) {
    // D = A(16x4) * B(4x16) + C, fp32 -> v_wmma_f32_16x16x4_f32
    return __builtin_amdgcn_wmma_f32_16x16x4_f32(false, a, false, b, (short)0, c,
                                                 false, false);
}

// ---------------------------------------------------------------------------
// Problem constants
// ---------------------------------------------------------------------------
#define NB   8          // batch
#define NC   32         // channels in/out
#define HH   256
#define WW   256
#define KTAP 49         // 7x7
#define KPAD 52         // padded to multiple of 4 for x4 WMMA k-steps
#define NROW 1024       // (o,i) pairs
#define PH   262        // padded FFT size
#define PX   16         // pixels per workgroup
#define NWG  4096       // 256 rows * 16 col-tiles

// Workspace float offsets
#define WS_WRE   0
#define WS_WIM   (NROW * KPAD)                 // 53248
#define WS_PART  (2 * NROW * KPAD)            // 106496  (4096 * 4 floats)
#define WS_SCL   (WS_PART + NWG * 4)          // 122880

// Dynamic LDS float offsets for the main kernel
#define L_TRE    0                            // 52*16
#define L_TIM    (52 * PX)                    // 832
#define L_SIG    (2 * 52 * PX)                // 1664 ; 16 px * 264 (8*33 padded)
#define L_KFR    (L_SIG + PX * 264)           // 5888 ; 16 px * 1056 (32*33 padded)
#define L_KFI    (L_KFR + PX * 1056)          // 22784
#define L_TOTAL  (L_KFI + PX * 1056)          // 39680 floats = 158720 bytes
#define L_OUTST  L_KFR                        // alias: 8*32*16*2 = 8192 floats

// ---------------------------------------------------------------------------
// Kernel 1: analytic Kf-std from weights (Parseval) + build swizzled,
// fftshift-permuted, scaled weight matrices W_re/W_im [1024][52].
// ---------------------------------------------------------------------------
__global__ void __launch_bounds__(256)
prep_weights(const float* __restrict__ wre_g, const float* __restrict__ wim_g,
             float* __restrict__ ws) {
    __shared__ float rq[256], rr[256], ri[256];
    __shared__ float sscale;
    const int tid = threadIdx.x;

    float ssq = 0.f, s0r = 0.f, s0i = 0.f;
    for (int idx = tid; idx < NC * NC * KTAP; idx += 256) {
        float a = wre_g[idx], b = wim_g[idx];
        ssq += a * a + b * b;
        if (idx % KTAP == 0) { s0r += a; s0i += b; }   // tap (kh=0,kw=0)
    }
    rq[tid] = ssq; rr[tid] = s0r; ri[tid] = s0i;
    __syncthreads();
    for (int s = 128; s > 0; s >>= 1) {
        if (tid < s) {
            rq[tid] += rq[tid + s];
            rr[tid] += rr[tid + s];
            ri[tid] += ri[tid + s];
        }
        __syncthreads();
    }
    if (tid == 0) {
        double P    = (double)PH * (double)PH;        // 262^2
        double N    = 1024.0 * P;                      // total Kf elements
        double sum2 = P * (double)rq[0];               // Parseval: sum |Kf|^2
        double mre  = P * (double)rr[0];               // sum Kf (complex)
        double mim  = P * (double)ri[0];
        double var  = (sum2 - (mre * mre + mim * mim) / N) / (N - 1.0);
        double sd   = sqrt(var > 0.0 ? var : 0.0);
        sscale = (float)(1.0 / (sd + 1e-10));
    }
    __syncthreads();
    const float sc = sscale;

    float* WR = ws + WS_WRE;
    float* WI = ws + WS_WIM;
    for (int idx = tid; idx < NROW * KPAD; idx += 256) {
        int row = idx / KPAD;
        int k   = idx - row * KPAD;
        float vr = 0.f, vi = 0.f;
        if (k < KTAP) {
            int o  = ((row >> 5) + 16) & 31;           // channel-dim fftshift
            int ii = ((row & 31) + 16) & 31;
            int src = (o * NC + ii) * KTAP + k;        // k == kh*7+kw already
            vr = wre_g[src] * sc;
            vi = wim_g[src] * sc;
        }
        WR[idx] = vr;
        WI[idx] = vi;
    }
}

// ---------------------------------------------------------------------------
// Kernel 2: fused twiddle-DFT GEMM + per-pixel channel contraction (WMMA f32)
// ---------------------------------------------------------------------------
__global__ void __launch_bounds__(256)
fftconv_main(const float* __restrict__ sig, const float* __restrict__ WR,
             const float* __restrict__ WI, float* __restrict__ out,
             float* __restrict__ partials) {
    extern __shared__ float smem[];
    float* Tre = smem + L_TRE;
    float* Tim = smem + L_TIM;
    float* S   = smem + L_SIG;
    float* KFR = smem + L_KFR;
    float* KFI = smem + L_KFI;
    float* OS  = smem + L_OUTST;   // aliases KFR after it is consumed
    float* red = smem;             // aliases T after it is consumed

    const int tid  = threadIdx.x;
    const int wg   = blockIdx.x;
    const int h    = wg >> 4;
    const int w0   = (wg & 15) << 4;
    const int lane = tid & 31;
    const int wv   = tid >> 5;
    const int colp = lane & 15;
    const int ksel = (lane < 16) ? 0 : 2;

    // -------- stage twiddle tile T[k][p] = exp(-2*pi*i*(u*kh + v_p*kw)/262)
    {
        int u = h + 131; if (u >= PH) u -= PH;         // spatial fftshift
        const float mtp = -6.28318530717958647692f / (float)PH;
        for (int idx = tid; idx < KPAD * PX; idx += 256) {
            int k = idx >> 4, p = idx & 15;
            float re = 0.f, im = 0.f;
            if (k < KTAP) {
                int kh = k / 7, kw = k - kh * 7;
                int v = w0 + p + 131; if (v >= PH) v -= PH;
                float ang = mtp * (float)(u * kh + v * kw);
                __sincosf(ang, &im, &re);
            }
            Tre[k * PX + p] = re;
            Tim[k * PX + p] = im;
        }
    }
    // -------- stage signal tile S[p][b*33 + i] (pad 33 vs bank conflicts)
    {
        int b  = tid >> 5;
        int ci = tid & 31;
        const float* gp = sig + ((size_t)(b * NC + ci) << 16) + h * WW + w0;
        __builtin_prefetch(gp, 0, 1);                  // global_prefetch_b8
        float4 q0 = ((const float4*)gp)[0];
        float4 q1 = ((const float4*)gp)[1];
        float4 q2 = ((const float4*)gp)[2];
        float4 q3 = ((const float4*)gp)[3];
        float v[16] = {q0.x, q0.y, q0.z, q0.w, q1.x, q1.y, q1.z, q1.w,
                       q2.x, q2.y, q2.z, q2.w, q3.x, q3.y, q3.z, q3.w};
        int soff = b * 33 + ci;
#pragma unroll
        for (int p = 0; p < 16; ++p) S[p * 264 + soff] = v[p];
    }
    __syncthreads();

    // -------- stage 1: Kf[(oi),p] = W(1024x52) * T(52x16), complex GEMM
    // 8 waves x 8 M-tiles; 13 k-steps of 4; 4 real WMMAs per step
    for (int j = 0; j < 8; ++j) {
        int m   = wv * 8 + j;
        int row = m * 16 + colp;                       // A-operand M row
        const float* ar = WR + row * KPAD;
        const float* ai = WI + row * KPAD;
        v8f accR = {0.f, 0.f, 0.f, 0.f, 0.f, 0.f, 0.f, 0.f};
        v8f accI = {0.f, 0.f, 0.f, 0.f, 0.f, 0.f, 0.f, 0.f};
        for (int ks = 0; ks < 13; ++ks) {
            int k0 = ks * 4 + ksel;
            v2f aR = *(const v2f*)(ar + k0);
            v2f aI = *(const v2f*)(ai + k0);
            v2f bR, bI;
            bR.x = Tre[k0 * PX + colp];
            bR.y = Tre[(k0 + 1) * PX + colp];
            bI.x = Tim[k0 * PX + colp];
            bI.y = Tim[(k0 + 1) * PX + colp];
            accR = wmma4(aR, bR, accR);                // +Wre*Tre
            v2f aIn; aIn.x = -aI.x; aIn.y = -aI.y;     // NEG bits are C-only
            accR = wmma4(aIn, bI, accR);               // -Wim*Tim
            accI = wmma4(aR, bI, accI);                // +Wre*Tim
            accI = wmma4(aI, bR, accI);                // +Wim*Tre
        }
        // D layout: col = pixel p = lane%16; vgpr r -> row r (lanes<16) / r+8
        int rbase = m * 16 + ((lane < 16) ? 0 : 8);
        float* kr = KFR + colp * 1056;                 // [p][o*33+i]
        float* ki = KFI + colp * 1056;
#pragma unroll
        for (int r = 0; r < 8; ++r) {
            int rr2 = rbase + r;
            int oo = rr2 >> 5, ii = rr2 & 31;
            kr[oo * 33 + ii] = accR[r];
            ki[oo * 33 + ii] = accI[r];
        }
    }
    __syncthreads();

    // -------- stage 2: out[b,o] = sum_i s[b,i] * Kn[o,i]  (per pixel)
    // M = b (8 valid of 16), K = i (32), N = o (2 tiles of 16); 2 pixels/wave
    v8f zz = {0.f, 0.f, 0.f, 0.f, 0.f, 0.f, 0.f, 0.f};
    v8f accRe[2][2] = {{zz, zz}, {zz, zz}};
    v8f accIm[2][2] = {{zz, zz}, {zz, zz}};
    {
        const bool act = (colp < 8);                   // b rows 8..15 are pad
        for (int pp = 0; pp < 2; ++pp) {
            int p = wv * 2 + pp;
            const float* Sp = S + p * 264 + colp * 33;
            const float* kr = KFR + p * 1056;
            const float* ki = KFI + p * 1056;
            for (int ks = 0; ks < 8; ++ks) {
                int i0 = ks * 4 + ksel;
                v2f a;
                a.x = act ? Sp[i0]     : 0.f;
                a.y = act ? Sp[i0 + 1] : 0.f;
                int c0 = colp * 33 + i0;               // o-tile 0: o = colp
                int c1 = (colp + 16) * 33 + i0;        // o-tile 1: o = colp+16
                v2f bR0, bR1, bI0, bI1;
                bR0.x = kr[c0]; bR0.y = kr[c0 + 1];
                bR1.x = kr[c1]; bR1.y = kr[c1 + 1];
                bI0.x = ki[c0]; bI0.y = ki[c0 + 1];
                bI1.x = ki[c1]; bI1.y = ki[c1 + 1];
                accRe[pp][0] = wmma4(a, bR0, accRe[pp][0]);
                accRe[pp][1] = wmma4(a, bR1, accRe[pp][1]);
                accIm[pp][0] = wmma4(a, bI0, accIm[pp][0]);
                accIm[pp][1] = wmma4(a, bI1, accIm[pp][1]);
            }
        }
    }
    __syncthreads();   // everyone done reading KFR/KFI -> reuse as OS

    // -------- park results in LDS (interleaved complex) for coalesced stores
    if (lane < 16) {
        for (int pp = 0; pp < 2; ++pp) {
            int p = wv * 2 + pp;
#pragma unroll
            for (int r = 0; r < 8; ++r) {              // b = r (valid rows)
                int base0 = (((r * NC + colp) * PX + p) << 1);
                int base1 = (((r * NC + colp + 16) * PX + p) << 1);
                OS[base0 + 0] = accRe[pp][0][r];
                OS[base0 + 1] = accIm[pp][0][r];
                OS[base1 + 0] = accRe[pp][1][r];
                OS[base1 + 1] = accIm[pp][1][r];
            }
        }
    }
    __syncthreads();

    // -------- coalesced global store + fixed-order partial std sums
    float sr = 0.f, si = 0.f, sq = 0.f;
    {
        const float* src = OS + tid * 32;              // (b,o) pair = tid
        float* dst = out + (((size_t)tid << 16) + (size_t)(h * WW + w0)) * 2;
#pragma unroll
        for (int q = 0; q < 8; ++q) {
            float4 vq = ((const float4*)src)[q];
            ((float4*)dst)[q] = vq;
            sr += vq.x + vq.z;
            si += vq.y + vq.w;
            sq += vq.x * vq.x + vq.y * vq.y + vq.z * vq.z + vq.w * vq.w;
        }
    }
    red[tid] = sr; red[256 + tid] = si; red[512 + tid] = sq;
    __syncthreads();
    for (int s = 128; s > 0; s >>= 1) {
        if (tid < s) {
            red[tid]       += red[tid + s];
            red[256 + tid] += red[256 + tid + s];
            red[512 + tid] += red[512 + tid + s];
        }
        __syncthreads();
    }
    if (tid == 0) {
        partials[wg * 4 + 0] = red[0];
        partials[wg * 4 + 1] = red[256];
        partials[wg * 4 + 2] = red[512];
    }
}

// ---------------------------------------------------------------------------
// Kernel 3: reduce per-WG partials -> 1/(std(out)+eps)
// ---------------------------------------------------------------------------
__global__ void __launch_bounds__(256)
finalize_stats(const float* __restrict__ partials, float* __restrict__ scl) {
    __shared__ float rr[256], ri[256], rq[256];
    const int tid = threadIdx.x;
    float a = 0.f, b = 0.f, c = 0.f;
    for (int i = tid; i < NWG; i += 256) {
        a += partials[i * 4 + 0];
        b += partials[i * 4 + 1];
        c += partials[i * 4 + 2];
    }
    rr[tid] = a; ri[tid] = b; rq[tid] = c;
    __syncthreads();
    for (int s = 128; s > 0; s >>= 1) {
        if (tid < s) {
            rr[tid] += rr[tid + s];
            ri[tid] += ri[tid + s];
            rq[tid] += rq[tid + s];
        }
        __syncthreads();
    }
    if (tid == 0) {
        double N   = (double)NB * NC * HH * WW;        // complex elements
        double var = ((double)rq[0] -
                      ((double)rr[0] * rr[0] + (double)ri[0] * ri[0]) / N) /
                     (N - 1.0);
        double sd = sqrt(var > 0.0 ? var : 0.0);
        scl[0] = (float)(1.0 / (sd + 1e-10));
    }
}

// ---------------------------------------------------------------------------
// Kernel 4: in-place rescale of out by the broadcast scalar (float4 stride)
// ---------------------------------------------------------------------------
__global__ void __launch_bounds__(256)
scale_out(float* __restrict__ out, const float* __restrict__ scl) {
    const float sc = scl[0];
    float4* o4 = (float4*)out;
    const size_t n4 = (size_t)NB * NC * HH * WW * 2 / 4;   // 8,388,608
    for (size_t i = blockIdx.x * 256 + threadIdx.x; i < n4;
         i += (size_t)4096 * 256) {
        float4 v = o4[i];
        v.x *= sc; v.y *= sc; v.z *= sc; v.w *= sc;
        o4[i] = v;
    }
}

// ---------------------------------------------------------------------------
extern "C" void kernel_launch(void* const* d_in, const int* in_sizes, int n_in,
                              void* d_out, int out_size, void* d_ws,
                              size_t ws_size, hipStream_t stream) {
    const float* sig = (const float*)d_in[0];
    const float* wre = (const float*)d_in[1];
    const float* wim = (const float*)d_in[2];
    float* out = (float*)d_out;
    float* ws  = (float*)d_ws;

    float* WR       = ws + WS_WRE;
    float* WI       = ws + WS_WIM;
    float* partials = ws + WS_PART;
    float* scl      = ws + WS_SCL;

    prep_weights<<<1, 256, 0, stream>>>(wre, wim, ws);
    fftconv_main<<<NWG, 256, L_TOTAL * sizeof(float), stream>>>(sig, WR, WI,
                                                                out, partials);
    finalize_stats<<<1, 256, 0, stream>>>(partials, scl);
    scale_out<<<4096, 256, 0, stream>>>(out, scl);
}